// MSE_Loss_74844100100229
// MI455X (gfx1250) — compile-verified
//
#include <hip/hip_runtime.h>

// 128-bit vector load type and WMMA operand types
typedef float  float4v __attribute__((ext_vector_type(4)));
typedef float  v2f     __attribute__((ext_vector_type(2)));
typedef float  v8f     __attribute__((ext_vector_type(8)));

// ---------------------------------------------------------------------------
// Wave32 reduction via V_WMMA_F32_16X16X4_F32.
//
// Each lane contributes two f32 accumulators (acc0, acc1). Per the CDNA5 ISA
// (CDNA5 ISA, "32-bit A-Matrix 16x4"):
//   lane m      (m<16): A[m][0] = v0, A[m][1] = v1
//   lane m+16         : A[m][2] = v0, A[m][3] = v1
// So (acc0, acc1) across the 32 lanes form exactly one 16x4 A matrix.
// With B = all-ones (an all-ones matrix is layout-invariant) and C = 0:
//   D[m][n] = sum_k A[m][k]   (row sums, replicated across n)
// D layout: VGPR r -> row r on lanes 0-15, row r+8 on lanes 16-31.
// Summing d[0..7] per lane gives rows 0-7 (lanes<16) / rows 8-15 (lanes>=16);
// one xor-16 shuffle produces the full wave sum in every lane. Exact f32 adds.
// ---------------------------------------------------------------------------
__device__ __forceinline__ float wave_reduce_wmma(float acc0, float acc1) {
    v2f a;
    a[0] = acc0;
    a[1] = acc1;
    v2f b;
    b[0] = 1.0f;
    b[1] = 1.0f;
    v8f c = {};
    // 8 args: (neg_a, A, neg_b, B, c_mod, C, reuse_a, reuse_b)
    v8f d = __builtin_amdgcn_wmma_f32_16x16x4_f32(
        /*neg_a=*/false, a, /*neg_b=*/false, b,
        /*c_mod=*/(short)0, c, /*reuse_a=*/false, /*reuse_b=*/false);
    float s = 0.0f;
#pragma unroll
    for (int r = 0; r < 8; ++r) s += d[r];
    // lanes 0-15 hold sum(rows 0-7), lanes 16-31 hold sum(rows 8-15)
    s += __shfl_xor(s, 16, 32);
    return s;  // full wave total in every lane
}

// Block-level finish: 8 waves (256 threads) -> one float.
__device__ __forceinline__ float block_reduce(float acc0, float acc1) {
    __shared__ float wsum[8];
    float w = wave_reduce_wmma(acc0, acc1);
    const int lane = threadIdx.x & 31;
    const int wave = threadIdx.x >> 5;
    if (lane == 0) wsum[wave] = w;
    __syncthreads();
    float s = 0.0f;
    if (threadIdx.x == 0) {
#pragma unroll
        for (int i = 0; i < 8; ++i) s += wsum[i];
    }
    return s;  // valid in thread 0 only
}

// ---------------------------------------------------------------------------
// Pass 1: streaming squared-difference partial sums.
// Fully coalesced 128-bit loads (global_load_b128) on both inputs; each
// element is touched exactly once -> bandwidth-bound at ~23.3 TB/s.
// ---------------------------------------------------------------------------
__global__ void __launch_bounds__(256)
mse_partial_kernel(const float* __restrict__ den, const float* __restrict__ img,
                   float* __restrict__ partial, long n4) {
    const float4v* __restrict__ a4 = (const float4v*)den;
    const float4v* __restrict__ b4 = (const float4v*)img;

    long tid    = (long)blockIdx.x * blockDim.x + threadIdx.x;
    long stride = (long)gridDim.x * blockDim.x;

    float acc0 = 0.0f, acc1 = 0.0f;
    for (long i = tid; i < n4; i += stride) {
        float4v a = a4[i];
        float4v b = b4[i];
        float d0 = a[0] - b[0];
        float d1 = a[1] - b[1];
        float d2 = a[2] - b[2];
        float d3 = a[3] - b[3];
        acc0 = __builtin_fmaf(d0, d0, acc0);
        acc1 = __builtin_fmaf(d1, d1, acc1);
        acc0 = __builtin_fmaf(d2, d2, acc0);
        acc1 = __builtin_fmaf(d3, d3, acc1);
    }

    float s = block_reduce(acc0, acc1);
    if (threadIdx.x == 0) partial[blockIdx.x] = s;
}

// ---------------------------------------------------------------------------
// Pass 2: single block reduces the per-block partials and produces
// out[0] = (total / (H*H) / (B*C))^2  (deterministic fixed-tree reduction).
// ---------------------------------------------------------------------------
__global__ void __launch_bounds__(256)
mse_final_kernel(const float* __restrict__ partial, float* __restrict__ out,
                 int n, float scale) {
    float acc0 = 0.0f, acc1 = 0.0f;
    for (int i = (int)threadIdx.x * 2; i < n; i += (int)blockDim.x * 2) {
        acc0 += partial[i];
        if (i + 1 < n) acc1 += partial[i + 1];
    }
    float s = block_reduce(acc0, acc1);
    if (threadIdx.x == 0) {
        float loss = s * scale;
        out[0] = loss * loss;
    }
}

// ---------------------------------------------------------------------------
// Launcher. Two-pass, no atomics -> bitwise deterministic across replays.
// ---------------------------------------------------------------------------
extern "C" void kernel_launch(void* const* d_in, const int* in_sizes, int n_in,
                              void* d_out, int out_size, void* d_ws, size_t ws_size,
                              hipStream_t stream) {
    const float* den = (const float*)d_in[0];  // denoised_img, [64,1,512,512] f32
    const float* img = (const float*)d_in[1];  // img,          [64,1,512,512] f32
    float*       out = (float*)d_out;          // scalar f32
    float*       ws  = (float*)d_ws;           // per-block partials

    const long n  = (long)in_sizes[0];         // 16,777,216 elements
    const long n4 = n >> 2;                    // float4 count (n divisible by 4)

    // Grid sizing: plenty of waves to cover HBM latency; partials fit in d_ws.
    int blocks = 1024;
    long max_partials = (long)(ws_size / sizeof(float));
    if (max_partials < blocks) blocks = (int)max_partials;
    if (blocks < 1) blocks = 1;

    mse_partial_kernel<<<blocks, 256, 0, stream>>>(den, img, ws, n4);

    // scale = 1 / (H*H * B*C) = 1 / (512*512 * 64) = 2^-24 (exact in f32)
    const float scale = 1.0f / 16777216.0f;
    mse_final_kernel<<<1, 256, 0, stream>>>(ws, out, blocks, scale);
}